// Net2_27968827031711
// MI455X (gfx1250) — compile-verified
//
#include <hip/hip_runtime.h>
#include <hip/hip_bf16.h>

typedef __attribute__((ext_vector_type(2))) float v2f;
typedef __attribute__((ext_vector_type(8))) float v8f;

#define N_NODES 50000
#define N_EDGES 800000
#define F_IN    128
#define D1      300
#define D2      100

#define D1_PAD  304   // ceil16(300)
#define D2_PAD  112   // ceil16(100)
#define F_PAD   128   // ceil16(128)

// ---------------- degree / normalization ----------------

__global__ void deg_init_kernel(float* __restrict__ deg, int n) {
    int i = blockIdx.x * blockDim.x + threadIdx.x;
    if (i < n) deg[i] = 1.0f;   // self-loop contributes 1
}

__global__ void deg_accum_kernel(const int* __restrict__ dst, float* __restrict__ deg, int e) {
    int i = blockIdx.x * blockDim.x + threadIdx.x;
    if (i < e) atomicAdd(&deg[dst[i]], 1.0f);
}

__global__ void deg_rsqrt_kernel(float* __restrict__ deg, int n) {
    int i = blockIdx.x * blockDim.x + threadIdx.x;
    if (i < n) deg[i] = rsqrtf(deg[i]);   // deg -> dis in place
}

// ---------------- weight transpose + zero-pad (both dims) ----------------
// WT[c][k] = (c < Nout && k < K) ? W[k][c] : 0   for c in [0,NoutPad), k in [0,Kpad)
__global__ void padT_kernel(const float* __restrict__ W, float* __restrict__ WT,
                            int K, int Kpad, int Nout, int NoutPad) {
    int i = blockIdx.x * blockDim.x + threadIdx.x;
    if (i >= Kpad * NoutPad) return;
    int c = i / Kpad;
    int k = i - c * Kpad;
    WT[i] = (c < Nout && k < K) ? W[(size_t)k * Nout + c] : 0.f;
}

// ---------------- WMMA f32 GEMM: C = act(A) @ W (+bias) ----------------
// Each wave: MT row-tiles x one 16-col tile. K processed in chunks of 4 WMMA
// steps (16 K-values) with two statically distinct register buffer sets:
// chunk c+1's 20 b64 loads issue while chunk c's 16 WMMAs execute.
// A: [M x Kpad] rows of stride Astride. WT: [NoutPad x Kpad] transposed+padded.
// C: [M x Cstride], real columns = Nout, pad columns zero-filled.
// Kpad % 16 == 0, M % 16 == 0.

template <bool RELU_IN, bool RELU_OUT, bool HAS_BIAS, int MT>
__global__ __launch_bounds__(128) void gemm_wmma_f32_kernel(
    const float* __restrict__ A, int Astride,
    const float* __restrict__ WT, int Kpad,
    const float* __restrict__ bias,
    float* __restrict__ C, int Cstride,
    int M, int Nout) {
    const int lane = threadIdx.x & 31;
    const int wave = threadIdx.x >> 5;

    const int tileCol = blockIdx.x * 16;                    // < NoutPad always
    const int rowBase = (blockIdx.y * 4 + wave) * (MT * 16);

    const int  half = lane & 15;
    const bool hi   = lane >= 16;
    const int  koff = hi ? 2 : 0;      // K placement in A/B frag per ISA layout
    const int  bcol = tileCol + half;  // always valid in padded WT

    const float* Bp = WT + (size_t)bcol * Kpad + koff;

    const float* Ap[MT];
#pragma unroll
    for (int t = 0; t < MT; ++t) {
        int r = rowBase + t * 16 + half;
        if (r > M - 1) r = M - 1;      // clamp loads; stores guarded instead
        Ap[t] = A + (size_t)r * Astride + koff;
    }

    v8f acc[MT];
#pragma unroll
    for (int t = 0; t < MT; ++t) acc[t] = (v8f){0.f,0.f,0.f,0.f,0.f,0.f,0.f,0.f};

    // two named buffer sets (no rotation copies -> no register coalescing chain)
    v2f b0[4], a0[4][MT];
    v2f b1[4], a1[4][MT];

    auto load0 = [&](int kb) {
#pragma unroll
        for (int s = 0; s < 4; ++s) {
            b0[s] = *(const v2f*)(Bp + kb + 4 * s);
#pragma unroll
            for (int t = 0; t < MT; ++t) a0[s][t] = *(const v2f*)(Ap[t] + kb + 4 * s);
        }
    };
    auto load1 = [&](int kb) {
#pragma unroll
        for (int s = 0; s < 4; ++s) {
            b1[s] = *(const v2f*)(Bp + kb + 4 * s);
#pragma unroll
            for (int t = 0; t < MT; ++t) a1[s][t] = *(const v2f*)(Ap[t] + kb + 4 * s);
        }
    };
    auto mma0 = [&]() {
#pragma unroll
        for (int s = 0; s < 4; ++s)
#pragma unroll
            for (int t = 0; t < MT; ++t) {
                v2f a = a0[s][t];
                if (RELU_IN) { a.x = fmaxf(a.x, 0.f); a.y = fmaxf(a.y, 0.f); }
                acc[t] = __builtin_amdgcn_wmma_f32_16x16x4_f32(
                    false, a, false, b0[s], (short)0, acc[t], false, false);
            }
    };
    auto mma1 = [&]() {
#pragma unroll
        for (int s = 0; s < 4; ++s)
#pragma unroll
            for (int t = 0; t < MT; ++t) {
                v2f a = a1[s][t];
                if (RELU_IN) { a.x = fmaxf(a.x, 0.f); a.y = fmaxf(a.y, 0.f); }
                acc[t] = __builtin_amdgcn_wmma_f32_16x16x4_f32(
                    false, a, false, b1[s], (short)0, acc[t], false, false);
            }
    };

    const int nch = Kpad >> 4;          // chunks of 16 K-values
    load0(0);
    int c = 0;
    for (; c + 2 < nch; c += 2) {
        load1((c + 1) << 4);            // prefetch chunk c+1
        mma0();                         // consume chunk c
        load0((c + 2) << 4);            // prefetch chunk c+2
        mma1();                         // consume chunk c+1
    }
    if (c + 2 == nch) { load1((c + 1) << 4); mma0(); mma1(); }
    else              { mma0(); }       // c == nch-1

    // ---- epilogue: bias, activation, store (pad columns get zero) ----
    const bool colOK = bcol < Nout;
    if (bcol >= Cstride) return;
    const float bv = (HAS_BIAS && colOK) ? bias[bcol] : 0.f;

#pragma unroll
    for (int t = 0; t < MT; ++t) {
        const int tr = rowBase + t * 16;
        if (tr >= M) break;             // M % 16 == 0: tile all-or-nothing
        const int rbase = tr + (hi ? 8 : 0);
#pragma unroll
        for (int r = 0; r < 8; ++r) {
            float v = acc[t][r] + bv;
            if (RELU_OUT) v = fmaxf(v, 0.f);
            if (!colOK) v = 0.f;        // zero-fill pad columns of C
            C[(size_t)(rbase + r) * Cstride + bcol] = v;
        }
    }
}

// ---------------- aggregation ----------------

// agg has row stride D1_PAD; cols >= D1 zero-filled.
// agg[i,j] = dis[i]^2 * h[i,j] + bg[j]
__global__ void agg_init_kernel(const float* __restrict__ h, const float* __restrict__ dis,
                                const float* __restrict__ bg, float* __restrict__ agg,
                                int total) {
    int i = blockIdx.x * blockDim.x + threadIdx.x;
    if (i >= total) return;
    int row = i / D1_PAD;
    int col = i - row * D1_PAD;
    float v = 0.f;
    if (col < D1) {
        float d = dis[row];
        v = d * d * h[(size_t)row * D1 + col] + bg[col];
    }
    agg[i] = v;
}

// agg[dst] += dis[src]*dis[dst] * h[src]; one thread per (edge, float4 chunk)
__global__ void scatter_kernel(const int* __restrict__ src, const int* __restrict__ dst,
                               const float* __restrict__ dis, const float* __restrict__ h,
                               float* __restrict__ agg, int e) {
    const int CH = D1 / 4;  // 75 float4 chunks per edge
    int i = blockIdx.x * blockDim.x + threadIdx.x;
    if (i >= e * CH) return;
    int ed = i / CH;
    int q  = (i - ed * CH) * 4;
    int s = src[ed];
    int d = dst[ed];
    float w = dis[s] * dis[d];
    float4 v = *reinterpret_cast<const float4*>(h + (size_t)s * D1 + q);
    float* ap = agg + (size_t)d * D1_PAD + q;
    atomicAdd(ap + 0, w * v.x);
    atomicAdd(ap + 1, w * v.y);
    atomicAdd(ap + 2, w * v.z);
    atomicAdd(ap + 3, w * v.w);
}

// ---------------- launch ----------------

extern "C" void kernel_launch(void* const* d_in, const int* in_sizes, int n_in,
                              void* d_out, int out_size, void* d_ws, size_t ws_size,
                              hipStream_t stream) {
    const float* x  = (const float*)d_in[0];
    const int*   ei = (const int*)d_in[1];     // [2, E]: src row then dst row
    const float* Wg = (const float*)d_in[2];
    const float* bg = (const float*)d_in[3];
    const float* W1 = (const float*)d_in[4];
    const float* b1 = (const float*)d_in[5];
    const float* W2 = (const float*)d_in[6];
    const float* b2 = (const float*)d_in[7];
    float* out = (float*)d_out;

    const int* src = ei;
    const int* dst = ei + N_EDGES;

    // workspace layout (floats)
    float* ws  = (float*)d_ws;
    float* dis = ws;                                   // N
    float* h   = dis + N_NODES;                        // N*D1        (60 MB)
    float* agg = h   + (size_t)N_NODES * D1;           // N*D1_PAD    (60.8 MB)
    float* h1  = agg + (size_t)N_NODES * D1_PAD;       // N*D2_PAD    (22.4 MB)
    float* WgT = h1  + (size_t)N_NODES * D2_PAD;       // D1_PAD*F_IN
    float* W1T = WgT + (size_t)D1_PAD * F_IN;          // D2_PAD*D1_PAD
    float* W2T = W1T + (size_t)D2_PAD * D1_PAD;        // F_PAD*D2_PAD

    const int TB = 256;
    const int ROWS_PER_BLK = 4 /*waves*/ * 4 /*MT*/ * 16;   // 256
    const int GRID_Y = (N_NODES + ROWS_PER_BLK - 1) / ROWS_PER_BLK;

    // 0) transpose + pad weights (tiny)
    padT_kernel<<<((F_IN   * D1_PAD) + TB - 1) / TB, TB, 0, stream>>>(Wg, WgT, F_IN, F_IN,   D1,   D1_PAD);
    padT_kernel<<<((D1_PAD * D2_PAD) + TB - 1) / TB, TB, 0, stream>>>(W1, W1T, D1,   D1_PAD, D2,   D2_PAD);
    padT_kernel<<<((D2_PAD * F_PAD)  + TB - 1) / TB, TB, 0, stream>>>(W2, W2T, D2,   D2_PAD, F_IN, F_PAD);

    // 1) degree (with self-loop) -> dis = deg^{-1/2}
    deg_init_kernel<<<(N_NODES + TB - 1) / TB, TB, 0, stream>>>(dis, N_NODES);
    deg_accum_kernel<<<(N_EDGES + TB - 1) / TB, TB, 0, stream>>>(dst, dis, N_EDGES);
    deg_rsqrt_kernel<<<(N_NODES + TB - 1) / TB, TB, 0, stream>>>(dis, N_NODES);

    // 2) h = x @ Wg       (M=50000, Kpad=128, Nout=300, C stride 300)
    {
        dim3 grid(D1_PAD / 16, GRID_Y);
        gemm_wmma_f32_kernel<false, false, false, 4><<<grid, 128, 0, stream>>>(
            x, F_IN, WgT, F_IN, nullptr, h, D1, N_NODES, D1);
    }

    // 3) agg = dis^2 * h + bg   (stride D1_PAD, pad cols zeroed)
    {
        int total = N_NODES * D1_PAD;
        agg_init_kernel<<<(total + TB - 1) / TB, TB, 0, stream>>>(h, dis, bg, agg, total);
    }

    // 4) agg[dst] += dis[src]*dis[dst]*h[src]
    {
        int total = N_EDGES * (D1 / 4);
        scatter_kernel<<<(total + TB - 1) / TB, TB, 0, stream>>>(src, dst, dis, h, agg, N_EDGES);
    }

    // 5) h1 = relu(relu(agg) @ W1 + b1)  (Kpad=304, Nout=100, C stride 112, pad zeroed)
    {
        dim3 grid(D2_PAD / 16, GRID_Y);
        gemm_wmma_f32_kernel<true, true, true, 4><<<grid, 128, 0, stream>>>(
            agg, D1_PAD, W1T, D1_PAD, b1, h1, D2_PAD, N_NODES, D2);
    }

    // 6) out = h1 @ W2 + b2              (Kpad=112, Nout=128, C stride 128)
    {
        dim3 grid(F_PAD / 16, GRID_Y);
        gemm_wmma_f32_kernel<false, false, true, 4><<<grid, 128, 0, stream>>>(
            h1, D2_PAD, W2T, D2_PAD, b2, out, F_IN, N_NODES, F_IN);
    }
}